// ParamsModel_34686155882534
// MI455X (gfx1250) — compile-verified
//
#include <hip/hip_runtime.h>
#include <stdint.h>

#define PEND_G  10.0f
#define T_STEPS 100
#define LANE_STRIDE_F 12   // floats per pendulum LDS row: 8 used + 4 pad = 48B (16B aligned, ~2-way bank conflicts)

// One RK4 "alt" step exactly matching the reference formulation.
__device__ __forceinline__ void rk4_alt_step(float& th, float& om, float a, float dt) {
  const float third = 1.0f / 3.0f;
  float k1t = om;
  float k1o = a * __sinf(th);
  float y2t = th + dt * k1t * third;
  float y2o = om + dt * k1o * third;
  float k2t = y2o;
  float k2o = a * __sinf(y2t);
  float y3t = th + dt * (k2t - k1t * third);
  float y3o = om + dt * (k2o - k1o * third);
  float k3t = y3o;
  float k3o = a * __sinf(y3t);
  float y4t = th + dt * (k1t - k2t + k3t);
  float y4o = om + dt * (k1o - k2o + k3o);
  float k4t = y4o;
  float k4o = a * __sinf(y4t);
  th += dt * (k1t + 3.0f * (k2t + k3t) + k4t) * 0.125f;
  om += dt * (k1o + 3.0f * (k2o + k3o) + k4o) * 0.125f;
}

// Robust LDS byte-address (workgroup-relative) via addrspacecast generic->local.
__device__ __forceinline__ unsigned lds_byte_addr(const void* p) {
  return (unsigned)(unsigned long long)(__attribute__((address_space(3))) const void*)p;
}

__global__ __launch_bounds__(256) void pendulum_rk4_kernel(
    const float* __restrict__ t, const float* __restrict__ z0,
    const float* __restrict__ params, float* __restrict__ out, int N) {
  // [wave][double-buffer][pendulum-lane][row]: 8*2*32*12*4B = 24 KB of the 320 KB WGP LDS
  __shared__ float buf[8][2][32][LANE_STRIDE_F];

  const int tid  = (int)threadIdx.x;
  const int wv   = tid >> 5;
  const int lane = tid & 31;
  const int wave_base = (int)(blockIdx.x * blockDim.x) + (wv << 5);
  const int pg = wave_base + lane;

  float th = 0.0f, om = 0.0f, a = 0.0f;
  if (pg < N) {
    th = z0[2 * pg + 0];
    om = z0[2 * pg + 1];
    a  = -PEND_G / params[pg];   // L == params, f = (omega, -(G/L) sin(theta))
  }

  if (wave_base + 32 <= N) {
    // ---- full wave: LDS-staged, async-DMA write path (N % 32 == 0, always taken) ----
    const unsigned long long out_base = (unsigned long long)(uintptr_t)out;
    const unsigned lds_wave = lds_byte_addr(&buf[wv][0][0][0]);

    for (int c = 0; c < 25; ++c) {          // 25 chunks x 4 time points = 100 outputs
      const int b = c & 1;                  // double buffer
      if (c >= 2) {
        // ensure the async stores that last read buffer b (chunk c-2) have drained
        asm volatile("s_wait_asynccnt 0x2" ::: "memory");
      }
      float* myrow = &buf[wv][b][lane][0];
#pragma unroll
      for (int j = 0; j < 4; ++j) {
        const int k = c * 4 + j;            // k==0 is the initial state y0
        if (k > 0) {
          const float dt = t[k] - t[k - 1];
          rk4_alt_step(th, om, a, dt);
        }
        myrow[2 * j + 0] = th;
        myrow[2 * j + 1] = om;
      }
      // LDS writes must land before the async engine reads them
      asm volatile("s_wait_dscnt 0x0" ::: "memory");
#pragma unroll
      for (int r = 0; r < 2; ++r) {
        // 64 (pendulum, 16B-half) pairs across 2 wave-wide async stores:
        // each instruction moves 32 lanes x 16B = 512B LDS -> global.
        const int idx = (r << 5) + lane;
        const int p = idx >> 1;             // pendulum within the wave
        const int q = idx & 1;              // which 16B half of the 32B chunk
        const unsigned lds_a = lds_wave +
            (unsigned)(b * (32 * LANE_STRIDE_F * 4) + p * (LANE_STRIDE_F * 4) + q * 16);
        const unsigned long long ga = out_base +
            (unsigned long long)((unsigned)(wave_base + p) * 800u +  // row stride T*2*4
                                 (unsigned)(c * 32 + q * 16));       // k0*8 + q*16
        asm volatile("global_store_async_from_lds_b128 %0, %1, off"
                     :: "v"(ga), "v"(lds_a)
                     : "memory");
      }
    }
    asm volatile("s_wait_asynccnt 0x0" ::: "memory");
  } else if (pg < N) {
    // ---- partial-wave fallback (never hit for N=200000): direct stores ----
    float* o = out + (size_t)pg * (T_STEPS * 2);
    o[0] = th;
    o[1] = om;
    for (int k = 1; k < T_STEPS; ++k) {
      const float dt = t[k] - t[k - 1];
      rk4_alt_step(th, om, a, dt);
      o[2 * k + 0] = th;
      o[2 * k + 1] = om;
    }
  }
}

extern "C" void kernel_launch(void* const* d_in, const int* in_sizes, int n_in,
                              void* d_out, int out_size, void* d_ws, size_t ws_size,
                              hipStream_t stream) {
  (void)n_in; (void)out_size; (void)d_ws; (void)ws_size;
  // setup_inputs order: mini_batch (unused), t, z0, params
  const float* t      = (const float*)d_in[1];
  const float* z0     = (const float*)d_in[2];
  const float* params = (const float*)d_in[3];
  float* out = (float*)d_out;
  const int N = in_sizes[3];                // params has N elements

  dim3 block(256);                          // 8 wave32s per WGP-resident block
  dim3 grid((unsigned)((N + 255) / 256));
  hipLaunchKernelGGL(pendulum_rk4_kernel, grid, block, 0, stream, t, z0, params, out, N);
}